// CriticModel_5677946765955
// MI455X (gfx1250) — compile-verified
//
#include <hip/hip_runtime.h>
#include <hip/hip_bf16.h>

typedef float v2f __attribute__((ext_vector_type(2)));
typedef float v8f __attribute__((ext_vector_type(8)));

__device__ __forceinline__ void atomAddF(float* p, float v) {
  unsafeAtomicAdd(p, v);   // native global_atomic_add_f32 on gfx1250
}

// ---------------------------------------------------------------------------
// C[M,NOUT] = A[M,K] @ B[K,NOUT] (+ bias), fp32 via V_WMMA_F32_16X16X4_F32.
// 256 threads = 8 waves; each wave computes a 16-row stripe across all NOUT
// columns (NOUT/16 accumulators). B staged through LDS in 32-row chunks,
// k-pair packed as float2 so each B fragment is one ds_load_b64; pair-row
// stride padded so lane halves hit disjoint LDS bank halves.
// ---------------------------------------------------------------------------
template <int K, int NOUT>
__global__ __launch_bounds__(256) void gemm_bias_wmma(
    const float* __restrict__ A, const float* __restrict__ B,
    const float* __restrict__ bias, float* __restrict__ C, int M)
{
  constexpr int KC   = 32;          // k-chunk staged in LDS
  constexpr int NPF2 = NOUT + 16;   // pair-row stride in float2 units:
                                    // lane-half offset = 2*NPF2 dwords == 32 mod 64
  __shared__ float Blds[(KC / 2) * NPF2 * 2];

  const int tid  = threadIdx.x;
  const int lane = tid & 31;
  const int wave = tid >> 5;
  const int lh   = lane >> 4;       // lane half: selects K pair / M+8
  const int lm   = lane & 15;

  const long m0 = (long)blockIdx.x * 128 + wave * 16;
  long mrow = m0 + lm;
  if (mrow >= M) mrow = M - 1;      // clamp loads; stores are guarded
  const float* Ap = A + mrow * (long)K + 2 * lh;

  v8f acc[NOUT / 16] = {};

#pragma unroll 1
  for (int k0 = 0; k0 < K; k0 += KC) {
    // stage B[k0:k0+KC][0:NOUT] k-pair-packed: Blds2[kr/2][c] = {B[kr][c], B[kr+1][c]}
    for (int idx = tid * 4; idx < KC * NOUT; idx += 256 * 4) {
      const int kr = idx / NOUT, c = idx % NOUT;
      const float4 v = *(const float4*)(B + (long)(k0 + kr) * NOUT + c);
      const int base = ((kr >> 1) * NPF2 + c) * 2 + (kr & 1);
      Blds[base + 0] = v.x; Blds[base + 2] = v.y;
      Blds[base + 4] = v.z; Blds[base + 6] = v.w;
    }
    __syncthreads();
#pragma unroll
    for (int kk = 0; kk < KC; kk += 4) {
      const v2f a  = *(const v2f*)(Ap + k0 + kk);  // A frag: M=lane%16, K=kk+2*lh+{0,1}
      const int r2 = (kk >> 1) + lh;               // k-pair row for this lane half
      v2f bf[NOUT / 16];
#pragma unroll
      for (int t = 0; t < NOUT / 16; ++t)          // batch: 16x ds_load_b64
        bf[t] = *(const v2f*)(Blds + ((r2 * NPF2 + t * 16 + lm) << 1));
#pragma unroll
      for (int t = 0; t < NOUT / 16; ++t)          // dense WMMA burst
        acc[t] = __builtin_amdgcn_wmma_f32_16x16x4_f32(
            false, a, false, bf[t], (short)0, acc[t], false, false);
    }
    __syncthreads();
  }

#pragma unroll
  for (int t = 0; t < NOUT / 16; ++t) {
    const int col = t * 16 + lm;
    const float bv = bias ? bias[col] : 0.0f;
#pragma unroll
    for (int r = 0; r < 8; ++r) {                  // D: VGPR r -> M = 8*lh + r
      const long row = m0 + 8 * lh + r;
      if (row < M) C[row * (long)NOUT + col] = acc[t][r] + bv;
    }
  }
}

// ---------------------------------------------------------------------------
// mean(edge_attr) partial sums: 64 blocks x 256 threads, 256 edge-strided groups
// ---------------------------------------------------------------------------
__global__ __launch_bounds__(256) void edge_mean_partial(
    const float* __restrict__ eattr, float* __restrict__ msum, int E)
{
  const int c   = threadIdx.x & 63;
  const int grp = blockIdx.x * 4 + (threadIdx.x >> 6);
  float s = 0.0f;
  for (long e = grp; e < E; e += 256) s += eattr[e * 64 + c];
  atomAddF(&msum[c], s);
}

// eloop[n] = (mean edge_attr) @ We_l   (self-loop edge projection, 256 vals)
__global__ void eloop_kernel(const float* __restrict__ msum,
                             const float* __restrict__ We_l,
                             float* __restrict__ eloop, float invE)
{
  const int n = threadIdx.x;
  float s = 0.0f;
  for (int k = 0; k < 64; ++k) s = fmaf(msum[k] * invE, We_l[k * 256 + n], s);
  eloop[n] = s;
}

// ---------------------------------------------------------------------------
// One wave per edge: m = leaky_relu(xl[src]+xr[dst]+e); logits = m . att
// segment-max into lmax via ordered-uint atomicMax
// ---------------------------------------------------------------------------
__global__ __launch_bounds__(256) void edge_logits_kernel(
    const float* __restrict__ xl, const float* __restrict__ xr,
    const float* __restrict__ eproj, const float* __restrict__ eloop,
    const int* __restrict__ src, const int* __restrict__ dst,
    const float* __restrict__ att,
    float* __restrict__ logits, unsigned int* __restrict__ lmax,
    int E, int Etot)
{
  const long e = (long)blockIdx.x * 8 + (threadIdx.x >> 5);
  if (e >= Etot) return;
  const int lane = threadIdx.x & 31;
  int s, d; const float* ep;
  if (e < E) { s = src[e]; d = dst[e]; ep = eproj + e * 256; }
  else       { s = d = (int)(e - E);   ep = eloop; }
  const int c0 = lane * 8;                       // 8 channels/lane, head = lane/8
  const float* pa = xl + (long)s * 256 + c0;
  const float* pb = xr + (long)d * 256 + c0;
  float part = 0.0f;
#pragma unroll
  for (int i = 0; i < 8; i += 4) {
    const float4 a = *(const float4*)(pa + i);
    const float4 b = *(const float4*)(pb + i);
    const float4 c = *(const float4*)(ep + c0 + i);
    float m0v = a.x + b.x + c.x; m0v = m0v > 0.0f ? m0v : 0.2f * m0v;
    float m1v = a.y + b.y + c.y; m1v = m1v > 0.0f ? m1v : 0.2f * m1v;
    float m2v = a.z + b.z + c.z; m2v = m2v > 0.0f ? m2v : 0.2f * m2v;
    float m3v = a.w + b.w + c.w; m3v = m3v > 0.0f ? m3v : 0.2f * m3v;
    part += m0v * att[c0 + i]     + m1v * att[c0 + i + 1]
          + m2v * att[c0 + i + 2] + m3v * att[c0 + i + 3];
  }
  part += __shfl_xor(part, 1);                   // reduce the 8 lanes of a head
  part += __shfl_xor(part, 2);
  part += __shfl_xor(part, 4);
  if ((lane & 7) == 0) {
    const int h = lane >> 3;
    logits[e * 4 + h] = part;
    unsigned int u = __float_as_uint(part);
    unsigned int key = (u & 0x80000000u) ? ~u : (u | 0x80000000u);
    atomicMax(&lmax[(long)d * 4 + h], key);      // init 0 < key(-inf); self-loop
  }                                              // guarantees every dst updated
}

// p = exp(logit - lmax[dst]); denom[dst] += p  (p overwrites logits in place)
__global__ __launch_bounds__(256) void softmax_p_kernel(
    const int* __restrict__ dst, float* __restrict__ pbuf,
    const unsigned int* __restrict__ lmax, float* __restrict__ denom,
    int E, int Etot)
{
  const long idx = (long)blockIdx.x * 256 + threadIdx.x;
  if (idx >= (long)Etot * 4) return;
  const long e = idx >> 2;
  const int  h = (int)(idx & 3);
  const int  d = (e < E) ? dst[e] : (int)(e - E);
  const unsigned int key = lmax[(long)d * 4 + h];
  const unsigned int u = (key & 0x80000000u) ? (key & 0x7fffffffu) : ~key;
  const float p = __expf(pbuf[idx] - __uint_as_float(u));
  pbuf[idx] = p;
  atomAddF(&denom[(long)d * 4 + h], p);
}

// acc[dst] += xl[src] * alpha   (one wave per edge, 8 f32 atomics per lane)
__global__ __launch_bounds__(256) void scatter_kernel(
    const float* __restrict__ xl, const float* __restrict__ pbuf,
    const float* __restrict__ denom,
    const int* __restrict__ src, const int* __restrict__ dst,
    float* __restrict__ acc, int E, int Etot)
{
  const long e = (long)blockIdx.x * 8 + (threadIdx.x >> 5);
  if (e >= Etot) return;
  const int lane = threadIdx.x & 31;
  int s, d;
  if (e < E) { s = src[e]; d = dst[e]; } else { s = d = (int)(e - E); }
  const int h = lane >> 3;
  const float alpha = pbuf[e * 4 + h] / (denom[(long)d * 4 + h] + 1e-16f);
  const int c0 = lane * 8;
  const float* xs = xl + (long)s * 256 + c0;
  float* ad = acc + (long)d * 256 + c0;
#pragma unroll
  for (int i = 0; i < 8; ++i) atomAddF(&ad[i], xs[i] * alpha);
}

__global__ __launch_bounds__(256) void bias_relu_kernel(
    const float* __restrict__ acc, const float* __restrict__ bias,
    float* __restrict__ out, long total)
{
  const long i = (long)blockIdx.x * 256 + threadIdx.x;
  if (i >= total) return;
  const float v = acc[i] + bias[(int)(i & 255)];
  out[i] = v > 0.0f ? v : 0.0f;
}

__global__ __launch_bounds__(256) void pool_kernel(
    const float* __restrict__ x, const int* __restrict__ batch,
    float* __restrict__ pool, float* __restrict__ cnt, long N)
{
  const long idx = (long)blockIdx.x * 256 + threadIdx.x;
  if (idx >= N * 256) return;
  const long node = idx >> 8;
  const int  c = (int)(idx & 255);
  const int  g = batch[node];
  atomAddF(&pool[(long)g * 256 + c], x[idx]);
  if (c == 0) atomAddF(&cnt[g], 1.0f);
}

// one block (128 threads) per graph: h = relu(pooled@Wp1+bp1); out = h@Wp2+bp2
__global__ __launch_bounds__(128) void predict_kernel(
    const float* __restrict__ pool, const float* __restrict__ cnt,
    const float* __restrict__ Wp1, const float* __restrict__ bp1,
    const float* __restrict__ Wp2, const float* __restrict__ bp2,
    float* __restrict__ out)
{
  const int g = blockIdx.x;
  const int t = threadIdx.x;
  const float invc = 1.0f / fmaxf(cnt[g], 1.0f);
  const float* pg = pool + (long)g * 256;
  float h = bp1[t];
  for (int k = 0; k < 256; ++k) h = fmaf(pg[k] * invc, Wp1[k * 128 + t], h);
  h = fmaxf(h, 0.0f);
  __shared__ float red[128];
  red[t] = h * Wp2[t];
  __syncthreads();
  for (int s = 64; s > 0; s >>= 1) {
    if (t < s) red[t] += red[t + s];
    __syncthreads();
  }
  if (t == 0) out[g] = red[0] + bp2[0];
}

// ---------------------------------------------------------------------------
extern "C" void kernel_launch(void* const* d_in, const int* in_sizes, int n_in,
                              void* d_out, int out_size, void* d_ws, size_t ws_size,
                              hipStream_t stream)
{
  const float* x     = (const float*)d_in[0];
  const int*   eidx  = (const int*)d_in[1];
  const float* eattr = (const float*)d_in[2];
  const int*   batch = (const int*)d_in[3];
  const float* Wl    = (const float*)d_in[4];
  const float* bl    = (const float*)d_in[5];
  const float* Wr    = (const float*)d_in[6];
  const float* br    = (const float*)d_in[7];
  const float* We    = (const float*)d_in[8];
  const float* att   = (const float*)d_in[9];
  const float* bias  = (const float*)d_in[10];
  const float* Wp1   = (const float*)d_in[11];
  const float* bp1   = (const float*)d_in[12];
  const float* Wp2   = (const float*)d_in[13];
  const float* bp2   = (const float*)d_in[14];

  const int N    = in_sizes[0] / 256;
  const int E    = in_sizes[1] / 2;
  const int G    = out_size;
  const int Etot = E + N;
  const int* src = eidx;
  const int* dst = eidx + E;

  char* w = (char*)d_ws;
  auto carve = [&](size_t bytes) -> char* {
    char* p = w; w += (bytes + 255) & ~(size_t)255; return p;
  };
  float*        xl     = (float*)carve((size_t)N * 256 * 4);
  float*        xr     = (float*)carve((size_t)N * 256 * 4);   // reused as acc
  float*        xbuf   = (float*)carve((size_t)N * 256 * 4);
  float*        eproj  = (float*)carve((size_t)E * 256 * 4);
  float*        logits = (float*)carve((size_t)Etot * 4 * 4);  // becomes p
  unsigned int* lmax   = (unsigned int*)carve((size_t)N * 4 * 4);
  float*        denom  = (float*)carve((size_t)N * 4 * 4);
  float*        msum   = (float*)carve(64 * 4);
  float*        eloop  = (float*)carve(256 * 4);
  float*        pool   = (float*)carve((size_t)G * 256 * 4);
  float*        cnt    = (float*)carve((size_t)G * 4);
  float*        acc    = xr;  // dead after edge_logits; reused for aggregation

  const unsigned gN  = (unsigned)((N + 127) / 128);
  const unsigned gE  = (unsigned)((E + 127) / 128);
  const unsigned gEd = (unsigned)((Etot + 7) / 8);
  const unsigned gL  = (unsigned)(((size_t)Etot * 4 + 255) / 256);
  const unsigned gNC = (unsigned)(((size_t)N * 256 + 255) / 256);

  hipMemsetAsync(msum, 0, 64 * 4, stream);
  edge_mean_partial<<<64, 256, 0, stream>>>(eattr, msum, E);

  const float* xcur = x;
  for (int l = 0; l < 3; ++l) {
    const float* Wl_l = Wl + (size_t)l * 256 * 256;
    const float* bl_l = bl + (size_t)l * 256;
    const float* Wr_l = Wr + (size_t)l * 256 * 256;
    const float* br_l = br + (size_t)l * 256;
    const float* We_l = We + (size_t)l * 64 * 256;
    const float* at_l = att + (size_t)l * 256;
    const float* bi_l = bias + (size_t)l * 256;

    gemm_bias_wmma<256, 256><<<gN, 256, 0, stream>>>(xcur, Wl_l, bl_l, xl, N);
    gemm_bias_wmma<256, 256><<<gN, 256, 0, stream>>>(xcur, Wr_l, br_l, xr, N);
    gemm_bias_wmma<64, 256><<<gE, 256, 0, stream>>>(eattr, We_l, nullptr, eproj, E);
    eloop_kernel<<<1, 256, 0, stream>>>(msum, We_l, eloop, 1.0f / (float)E);

    hipMemsetAsync(lmax, 0, (size_t)N * 16, stream);
    edge_logits_kernel<<<gEd, 256, 0, stream>>>(xl, xr, eproj, eloop, src, dst,
                                                at_l, logits, lmax, E, Etot);
    hipMemsetAsync(denom, 0, (size_t)N * 16, stream);
    softmax_p_kernel<<<gL, 256, 0, stream>>>(dst, logits, lmax, denom, E, Etot);

    hipMemsetAsync(acc, 0, (size_t)N * 256 * 4, stream);  // xr dead now
    scatter_kernel<<<gEd, 256, 0, stream>>>(xl, logits, denom, src, dst, acc, E, Etot);
    bias_relu_kernel<<<gNC, 256, 0, stream>>>(acc, bi_l, xbuf, (long)N * 256);
    xcur = xbuf;
  }

  hipMemsetAsync(pool, 0, (size_t)G * 256 * 4, stream);
  hipMemsetAsync(cnt, 0, (size_t)G * 4, stream);
  pool_kernel<<<gNC, 256, 0, stream>>>(xcur, batch, pool, cnt, N);
  predict_kernel<<<(unsigned)G, 128, 0, stream>>>(pool, cnt, Wp1, bp1, Wp2, bp2,
                                                  (float*)d_out);
}